// TwoTowerModel_65944927863321
// MI455X (gfx1250) — compile-verified
//
#include <hip/hip_runtime.h>
#include <hip/hip_bf16.h>

// ---------------------------------------------------------------------------
// Two-tower retrieval on MI455X (gfx1250, wave32):
//   K1: q[256][100] = user_table[ids] @ W + b           (trivial)
//   K2: fused  scores = q @ candidates^T  (fp32 WMMA 16x16x4)
//       - candidate tiles staged to LDS by the Tensor Data Mover
//         (tensor_load_to_lds, TENSORcnt), DOUBLE-BUFFERED so stage s+1
//         streams in while stage s is consumed
//       - B fragments via bank-conflict-free ds_load_b64
//       - split-K dual accumulator chains per wave
//       - exact per-chunk top-100 per query row (chunk = 2048, scores in LDS)
//   K3: merge 489 chunk-local top-100 lists per row -> global top-100
// Roofline: 51.2 GFLOP fp32, AI ~128 FLOP/B -> compute bound.
// LDS: 16*2048*4 (scores) + 2*128*100*4 (stage bufs) = 233.5 KB of 320 KB/WGP.
// ---------------------------------------------------------------------------

#define D_EMB    64
#define DS       100        // semb dim (= K of the GEMM)
#define KSEL     100        // top-k
#define CHUNK    2048       // candidates per workgroup in K2
#define STAGE    128        // candidates per TDM transfer (16 steps/chunk)
#define NSTAGE   (CHUNK/STAGE)
#define SEG      (CHUNK/32) // per-lane segment of the LDS score row = 64
#define SEG_SHIFT 6
#define SC_FLOATS  (16*CHUNK)         // score region, floats
#define STBUF_FLOATS (STAGE*DS)       // one stage buffer, floats (12800)
#define LDS_BYTES ((SC_FLOATS + 2*STBUF_FLOATS) * 4)

typedef float v2f __attribute__((ext_vector_type(2)));
typedef float v8f __attribute__((ext_vector_type(8)));
typedef unsigned int v4u __attribute__((ext_vector_type(4)));
typedef int v8i __attribute__((ext_vector_type(8)));
typedef int v4i __attribute__((ext_vector_type(4)));

// ---------------------------------------------------------------- kernel 1 --
__global__ __launch_bounds__(128) void user_tower_kernel(
    const int* __restrict__ uid, const float* __restrict__ table,
    const float* __restrict__ W, const float* __restrict__ bias,
    float* __restrict__ q)
{
  const int u = blockIdx.x;
  const int j = threadIdx.x;
  if (j >= DS) return;
  const float* e = table + (long)uid[u] * D_EMB;
  float acc = bias[j];
#pragma unroll
  for (int d = 0; d < D_EMB; ++d) acc += e[d] * W[d * DS + j];
  q[u * DS + j] = acc;
}

// ------------------------------------------------- TDM descriptor + issue ---
// 2-D tensor: row-major [nItems][DS] fp32; tile = rows x DS starting at gsrc.
// D# layout per CDNA5 ISA 8.3-8.5 (group0: count/lds/global/type,
// group1: data_size, tensor_dim0/1, tile_dim0/1, dim0_stride).
__device__ __forceinline__ void tdm_load_rows(const float* gsrc,
                                              unsigned ldsByte, int rows,
                                              int nItems)
{
  const unsigned long long ga = (unsigned long long)(uintptr_t)gsrc;
  v4u g0;
  g0.x = 1u;                                           // count=1 (valid, load)
  g0.y = ldsByte;                                      // lds_addr [63:32]
  g0.z = (unsigned)(ga & 0xffffffffu);                 // global_addr[31:0]
  g0.w = (unsigned)((ga >> 32) & 0x01ffffffu)          // global_addr[56:32]
         | (2u << 30);                                 // type=2 ("image")
  v8i g1;
  g1[0] = (int)(2u << 16);                             // data_size=2 (4 bytes)
  g1[1] = (int)((unsigned)DS << 16);                   // tensor_dim0[15:0]
  g1[2] = (int)(((unsigned)nItems & 0xffffu) << 16);   // dim0[31:16]=0|dim1 lo
  g1[3] = (int)((((unsigned)nItems >> 16) & 0xffffu)   // tensor_dim1[31:16]
                | ((unsigned)DS << 16));               // tile_dim0 = DS
  g1[4] = rows;                                        // tile_dim1 (dim2=0)
  g1[5] = DS;                                          // tensor_dim0_stride
  g1[6] = 0;                                           // stride hi | dim1_str
  g1[7] = 0;
  const v4i z4 = {0, 0, 0, 0};                         // groups 2/3: 2-D only
  const v8i z8 = {0, 0, 0, 0, 0, 0, 0, 0};
  // amdgpu-toolchain (clang-23) 6-arg form
  __builtin_amdgcn_tensor_load_to_lds(g0, g1, z4, z4, z8, 0);
}

// ---------------------------------------------------------------- kernel 2 --
__device__ __forceinline__ void build_top8(const float* rs, int lane,
                                           float ls[8], int li[8])
{
  const float NEGINF = -__builtin_inff();
#pragma unroll
  for (int i = 0; i < 8; ++i) { ls[i] = NEGINF; li[i] = -1; }
  const int segBase = lane * SEG;
  for (int j = 0; j < SEG; ++j) {
    // stagger by lane -> LDS bank = (j+lane)&63, conflict-free
    const int cidx = segBase + ((j + lane) & (SEG - 1));
    const float v = rs[cidx];
    if (v > ls[7]) {
      int p = 7;
      while (p > 0 && v > ls[p - 1]) { ls[p] = ls[p - 1]; li[p] = li[p - 1]; --p; }
      ls[p] = v; li[p] = cidx;
    }
  }
}

// grid: (nChunk, B/16), block: 256 threads = 8 waves.
__global__ __launch_bounds__(256) void score_topk_kernel(
    const float* __restrict__ q, const float* __restrict__ cand,
    int nItems, int nChunk,
    float* __restrict__ wsS, int* __restrict__ wsI)
{
  extern __shared__ float smem[];          // [scores 16xCHUNK][stage buf x2]
  float* stageBuf = smem + SC_FLOATS;
  const int chunk = blockIdx.x;
  const int mtile = blockIdx.y;
  const int tid   = threadIdx.x;
  const int lane  = tid & 31;
  const int wave  = tid >> 5;
  const int half  = lane >> 4;             // 0: lanes 0-15, 1: 16-31
  const int l16   = lane & 15;
  const float NEGINF = -__builtin_inff();

  // --- preload A fragments (q tile fixed per block) -----------------------
  // 32-bit A 16x4 layout: lane holds row M=lane&15, k-pair (4*kk+2*half, +1)
  v2f aF[25];
  {
    const float* qrow = q + (mtile * 16 + l16) * DS;
#pragma unroll
    for (int kk = 0; kk < 25; ++kk)
      aF[kk] = *(const v2f*)(qrow + kk * 4 + half * 2);
  }

  const int cbase0 = chunk * CHUNK;
  const unsigned stageByte0 = (unsigned)(uintptr_t)stageBuf;

  // --- prologue: TDM kick for stage 0 -------------------------------------
  if (wave == 0) {
    int rows0 = nItems - cbase0;
    if (rows0 > STAGE) rows0 = STAGE;
    tdm_load_rows(cand + (long)cbase0 * DS, stageByte0, rows0, nItems);
  }

  // --- 16 double-buffered stages of 128 candidates ------------------------
  for (int s = 0; s < NSTAGE; ++s) {
    if (wave == 0) {
      const int nxt = s + 1;
      bool issued = false;
      if (nxt < NSTAGE) {
        const int nb = cbase0 + nxt * STAGE;
        int rows = nItems - nb;
        if (rows > 0) {
          if (rows > STAGE) rows = STAGE;
          tdm_load_rows(cand + (long)nb * DS,
                        stageByte0 + (unsigned)(nxt & 1) * (STBUF_FLOATS * 4),
                        rows, nItems);
          issued = true;
        }
      }
      // TENSORcnt retires in order: <=1 keeps only stage s+1 in flight
      if (issued) __builtin_amdgcn_s_wait_tensorcnt(1);
      else        __builtin_amdgcn_s_wait_tensorcnt(0);
    }
    __syncthreads();                       // stage s visible to all waves

    // --- WMMA: 1 tile of 16 candidates per wave, split-K dual chains ------
    {
      const float* sb = stageBuf + (s & 1) * STBUF_FLOATS;
      const float* b0 = sb + (wave * 16 + l16) * DS;
      v8f acc0 = {}, acc1 = {};
#pragma unroll
      for (int kk = 0; kk < 24; kk += 2) {
        v2f f0 = *(const v2f*)(b0 + kk * 4 + half * 2);       // ds_load_b64
        v2f f1 = *(const v2f*)(b0 + (kk + 1) * 4 + half * 2);
        acc0 = __builtin_amdgcn_wmma_f32_16x16x4_f32(
            false, aF[kk], false, f0, (short)0, acc0, false, false);
        acc1 = __builtin_amdgcn_wmma_f32_16x16x4_f32(
            false, aF[kk + 1], false, f1, (short)0, acc1, false, false);
      }
      {
        v2f f0 = *(const v2f*)(b0 + 24 * 4 + half * 2);
        acc0 = __builtin_amdgcn_wmma_f32_16x16x4_f32(
            false, aF[24], false, f0, (short)0, acc0, false, false);
      }
      const int col0 = s * STAGE + wave * 16 + l16;   // local col in chunk
      const bool v0  = (cbase0 + col0) < nItems;
#pragma unroll
      for (int i = 0; i < 8; ++i) {        // D: VGPR i -> M = i + 8*half
        const int r = i + half * 8;
        smem[r * CHUNK + col0] = v0 ? (acc0[i] + acc1[i]) : NEGINF;
      }
    }
    __syncthreads();                       // buf (s&1) free for stage s+2
  }

  // --- exact per-chunk top-100: one wave per row, 2 rows per wave ---------
  for (int r = wave; r < 16; r += 8) {
    float* rs = smem + r * CHUNK;
    float ls[8]; int li[8];
    build_top8(rs, lane, ls, li);
    int popped = 0;
    const int outBase = ((mtile * 16 + r) * nChunk + chunk) * KSEL;

    for (int round = 0; round < KSEL; ++round) {
      if (popped == 8) {                   // lane cache drained: rescan
        build_top8(rs, lane, ls, li);      // extracted slots are -inf
        popped = 0;
      }
      float s = ls[popped];
      int   ix = li[popped];
#pragma unroll
      for (int off = 16; off > 0; off >>= 1) {  // wave32 butterfly argmax
        const float os = __shfl_xor(s, off, 32);
        const int   oi = __shfl_xor(ix, off, 32);
        if (os > s || (os == s && oi < ix)) { s = os; ix = oi; }
      }
      if (ix >= 0 && (ix >> SEG_SHIFT) == lane) {   // owner lane extracts
        rs[ix] = NEGINF;
        ++popped;
      }
      if (lane == 0) {
        wsS[outBase + round] = s;
        wsI[outBase + round] = (ix >= 0) ? (cbase0 + ix) : 0;
      }
    }
  }
}

// ---------------------------------------------------------------- kernel 3 --
__global__ __launch_bounds__(256) void merge_topk_kernel(
    float* __restrict__ wsS, const int* __restrict__ wsI, int nChunk,
    float* __restrict__ outS, int* __restrict__ outI)
{
  const int row = blockIdx.x;
  const int tid = threadIdx.x;
  const float NEGINF = -__builtin_inff();
  const int M    = nChunk * KSEL;
  const int base = row * M;
  const int seg  = (M + 255) >> 8;
  const int s0   = tid * seg;
  const int s1   = (s0 + seg < M) ? (s0 + seg) : M;

  __shared__ float redS[8];
  __shared__ int   redI[8];
  __shared__ float winS;
  __shared__ int   winI;

  float cs = NEGINF; int ci = -1;
  for (int j = s0; j < s1; ++j) { const float v = wsS[base + j]; if (v > cs) { cs = v; ci = j; } }

  for (int round = 0; round < KSEL; ++round) {
    float s = cs; int ix = ci;
#pragma unroll
    for (int off = 16; off > 0; off >>= 1) {
      const float os = __shfl_xor(s, off, 32);
      const int   oi = __shfl_xor(ix, off, 32);
      if (os > s || (os == s && oi < ix)) { s = os; ix = oi; }
    }
    if ((tid & 31) == 0) { redS[tid >> 5] = s; redI[tid >> 5] = ix; }
    __syncthreads();
    if (tid == 0) {
      float bs = redS[0]; int bi = redI[0];
#pragma unroll
      for (int w = 1; w < 8; ++w)
        if (redS[w] > bs || (redS[w] == bs && redI[w] < bi)) { bs = redS[w]; bi = redI[w]; }
      winS = bs; winI = bi;
    }
    __syncthreads();
    const float wsc = winS;
    const int   wi  = winI;
    if (wi >= s0 && wi < s1) {             // owner: extract + rescan segment
      wsS[base + wi] = NEGINF;
      cs = NEGINF; ci = -1;
      for (int j = s0; j < s1; ++j) { const float v = wsS[base + j]; if (v > cs) { cs = v; ci = j; } }
    }
    if (tid == 0) {
      outS[row * KSEL + round] = wsc;
      outI[row * KSEL + round] = (wi >= 0) ? wsI[base + wi] : 0;
    }
    __syncthreads();
  }
}

// ------------------------------------------------------------------ launch --
extern "C" void kernel_launch(void* const* d_in, const int* in_sizes, int n_in,
                              void* d_out, int out_size, void* d_ws, size_t ws_size,
                              hipStream_t stream) {
  const int*   user_ids = (const int*)d_in[0];
  const float* table    = (const float*)d_in[1];
  const float* W        = (const float*)d_in[2];
  const float* bias     = (const float*)d_in[3];
  const float* cand     = (const float*)d_in[4];

  const int B      = in_sizes[0];                    // 256
  const int nItems = in_sizes[4] / DS;               // 1,000,000
  const int nChunk = (nItems + CHUNK - 1) / CHUNK;   // 489

  // workspace: q [B][100] | wsS [B][nChunk][100] | wsI [B][nChunk][100]
  char*  ws  = (char*)d_ws;
  float* q   = (float*)ws;
  float* wsS = (float*)(ws + (size_t)B * DS * sizeof(float));
  int*   wsI = (int*)  (ws + (size_t)B * DS * sizeof(float)
                           + (size_t)B * nChunk * KSEL * sizeof(float));

  user_tower_kernel<<<B, 128, 0, stream>>>(user_ids, table, W, bias, q);

  dim3 g2(nChunk, B / 16);
  score_topk_kernel<<<g2, 256, LDS_BYTES, stream>>>(
      q, cand, nItems, nChunk, wsS, wsI);

  float* outS = (float*)d_out;
  int*   outI = (int*)d_out + B * KSEL;
  merge_topk_kernel<<<B, 256, 0, stream>>>(wsS, wsI, nChunk, outS, outI);
}